// ParametricGTCNN_Event_22050362098180
// MI455X (gfx1250) — compile-verified
//
#include <hip/hip_runtime.h>
#include <hip/hip_bf16.h>

// ---- problem constants (match reference) ----
constexpr int TT   = 12;     // obs window
constexpr int NN   = 2048;   // nodes
constexpr int BB   = 8;      // batch
constexpr int FH   = 64;     // hidden
constexpr int BT   = BB * TT;          // 96
constexpr int NBT  = NN * BT;          // 196608
constexpr int NF   = NN * FH;          // 131072
constexpr int BTNF = BT * NF;          // 12582912
constexpr int SAS  = 36;               // padded LDS row stride for S tiles (bank-conflict free)

// exp(-d/tau), tau=3
#define WD1 0.7165313106f
#define WD2 0.5134171190f
#define WD3 0.3678794412f

typedef float v2f __attribute__((ext_vector_type(2)));
typedef float v8f __attribute__((ext_vector_type(8)));
typedef int   v4i __attribute__((ext_vector_type(4)));

__device__ __forceinline__ v8f wmma4(v2f a, v2f b, v8f c) {
  // D = A(16x4,f32) x B(4x16,f32) + C(16x16,f32)
  return __builtin_amdgcn_wmma_f32_16x16x4_f32(false, a, false, b, (short)0, c,
                                               false, false);
}

// ---- global->LDS staging: async copy if the toolchain exposes it ----
#if __has_builtin(__builtin_amdgcn_global_load_async_to_lds_b128) && \
    __has_builtin(__builtin_amdgcn_s_wait_asynccnt)
#define HAVE_ASYNC_LDS 1
#else
#define HAVE_ASYNC_LDS 0
#endif

__device__ __forceinline__ void stage16(const float* __restrict__ g, float* l) {
#if HAVE_ASYNC_LDS
  __builtin_amdgcn_global_load_async_to_lds_b128(
      (__attribute__((address_space(1))) v4i*)(void*)g,
      (__attribute__((address_space(3))) v4i*)(void*)l, 0, 0);
#else
  *(float4*)l = *(const float4*)g;
#endif
}

__device__ __forceinline__ void stage_wait() {
#if HAVE_ASYNC_LDS
  __builtin_amdgcn_s_wait_asynccnt(0);
#endif
}

// ---------------------------------------------------------------------------
// x[b,0,n,t] -> H0[n, b*T+t]
__global__ void k_transpose_x(const float* __restrict__ x, float* __restrict__ H0) {
  int idx = blockIdx.x * 256 + threadIdx.x;          // < NBT
  int c = idx % BT, n = idx / BT;
  int b = c / TT, t = c % TT;
  H0[idx] = x[((size_t)b * NN + n) * TT + t];
}

// row sums of S
__global__ void k_rowsum(const float* __restrict__ S, float* __restrict__ rsS) {
  __shared__ float red[256];
  int m = blockIdx.x;
  float s = 0.f;
  for (int n = threadIdx.x; n < NN; n += 256) s += S[(size_t)m * NN + n];
  red[threadIdx.x] = s;
  __syncthreads();
  for (int st = 128; st > 0; st >>= 1) {
    if (threadIdx.x < st) red[threadIdx.x] += red[threadIdx.x + st];
    __syncthreads();
  }
  if (threadIdx.x == 0) rsS[m] = red[0];
}

// dinv[t,n] = 1/d where d>0 else 0
__global__ void k_dinv(const float* __restrict__ rsS, const float* __restrict__ sp,
                       float* __restrict__ dinv) {
  int idx = blockIdx.x * 256 + threadIdx.x;          // < T*N
  int t = idx / NN, n = idx % NN;
  float s0 = fmaxf(sp[0], 0.f), s1 = fmaxf(sp[1], 0.f);
  float s2 = fmaxf(sp[2], 0.f), s3 = fmaxf(sp[3], 0.f);
  float rsT = 0.f;
  if (t >= 1) rsT += WD1;
  if (t >= 2) rsT += WD2;
  if (t >= 3) rsT += WD3;
  float d = s0 + s1 * rsS[n] + s2 * rsT + s3 * rsT * rsS[n];
  dinv[idx] = d > 0.f ? 1.f / d : 0.f;
}

// layer-1 temporal mix: G[n,c] = s1*H + s3*sum_d wd*H[n,c-d]   (layout [n, 96])
__global__ void k_tmix1(const float* __restrict__ H, const float* __restrict__ sp,
                        float* __restrict__ G) {
  int idx = blockIdx.x * 256 + threadIdx.x;          // < NBT
  int t = (idx % BT) % TT;
  float s1 = fmaxf(sp[1], 0.f), s3 = fmaxf(sp[3], 0.f);
  float ht = 0.f;
  if (t >= 1) ht += WD1 * H[idx - 1];
  if (t >= 2) ht += WD2 * H[idx - 2];
  if (t >= 3) ht += WD3 * H[idx - 3];
  G[idx] = s1 * H[idx] + s3 * ht;
}

// layer-2 temporal mix on [bt, n, f]
__global__ void k_tmix2(const float* __restrict__ X, const float* __restrict__ sp,
                        float* __restrict__ G) {
  int idx = blockIdx.x * 256 + threadIdx.x;          // < BTNF
  int t = (idx / NF) % TT;
  float s1 = fmaxf(sp[1], 0.f), s3 = fmaxf(sp[3], 0.f);
  float ht = 0.f;
  if (t >= 1) ht += WD1 * X[idx - NF];
  if (t >= 2) ht += WD2 * X[idx - 2 * NF];
  if (t >= 3) ht += WD3 * X[idx - 3 * NF];
  G[idx] = s1 * X[idx] + s3 * ht;
}

// ---------------------------------------------------------------------------
// Layer-1 spatial GEMM + fused A_hat epilogue.
// Y[m,c] = (sum_n S[m,n]*G[n,c] + s0*H[m,c] + s2*HT[m,c]) * dinv[t,m]
// grid (N/128, 96/16). 8 waves/block, one 16x16 C-tile per wave.
__global__ void k_spat1(const float* __restrict__ S, const float* __restrict__ G,
                        const float* __restrict__ H, const float* __restrict__ dinv,
                        const float* __restrict__ sp, float* __restrict__ Y) {
  __shared__ float sA[128 * SAS];
  __shared__ float sB[32 * 16];
  const int mb = blockIdx.x * 128;
  const int cb = blockIdx.y * 16;
  const int tid = threadIdx.x;
  const int wave = tid >> 5, lane = tid & 31;
  const int l15 = lane & 15;
  const int hi = (lane >> 4) << 1;                   // 0 or 2
  const float s0 = fmaxf(sp[0], 0.f), s2 = fmaxf(sp[2], 0.f);
  v8f acc = {};
  for (int k0 = 0; k0 < NN; k0 += 32) {
    // stage S tile 128x32 (1024 float4, 4 per thread)
#pragma unroll
    for (int i = 0; i < 4; ++i) {
      int e4 = tid + i * 256, r = e4 >> 3, c4 = e4 & 7;
      stage16(&S[(size_t)(mb + r) * NN + k0 + 4 * c4], &sA[r * SAS + 4 * c4]);
    }
    // stage G tile 32x16 (128 float4, first 128 threads)
    if (tid < 128) {
      int r = tid >> 2, c4 = tid & 3;
      stage16(&G[(size_t)(k0 + r) * BT + cb + 4 * c4], &sB[r * 16 + 4 * c4]);
    }
    stage_wait();
    __syncthreads();
#pragma unroll
    for (int kk = 0; kk < 32; kk += 4) {
      int kb = kk + hi;
      v2f a;
      a.x = sA[(wave * 16 + l15) * SAS + kb];
      a.y = sA[(wave * 16 + l15) * SAS + kb + 1];
      v2f b;
      b.x = sB[kb * 16 + l15];
      b.y = sB[(kb + 1) * 16 + l15];
      acc = wmma4(a, b, acc);
    }
    __syncthreads();
  }
  const int c = cb + l15;
  const int t = c % TT;
  const int mbase = mb + wave * 16 + ((lane >> 4) << 3);
#pragma unroll
  for (int r = 0; r < 8; ++r) {
    int m = mbase + r;
    size_t hidx = (size_t)m * BT + c;
    float ht = 0.f;
    if (t >= 1) ht += WD1 * H[hidx - 1];
    if (t >= 2) ht += WD2 * H[hidx - 2];
    if (t >= 3) ht += WD3 * H[hidx - 3];
    Y[hidx] = (acc[r] + s0 * H[hidx] + s2 * ht) * dinv[t * NN + m];
  }
}

// Layer-1 output: X[bt,n,g] = relu(b1 + H0*W1[0] + H1*W1[1] + H2*W1[2])
__global__ void k_l1outer(const float* __restrict__ H0, const float* __restrict__ H1,
                          const float* __restrict__ H2, const float* __restrict__ W1,
                          const float* __restrict__ b1, float* __restrict__ X) {
  int idx = blockIdx.x * 256 + threadIdx.x;          // < BTNF
  int g = idx & 63;
  int rem = idx >> 6;                                // bt*N + n
  int n = rem & (NN - 1);
  int bt = rem >> 11;
  size_t h = (size_t)n * BT + bt;
  float v = b1[g] + H0[h] * W1[g] + H1[h] * W1[64 + g] + H2[h] * W1[128 + g];
  X[idx] = fmaxf(v, 0.f);
}

// ---------------------------------------------------------------------------
// Layer-2 spatial GEMM + fused A_hat epilogue on [bt, n, 64] state.
// grid (N/256, B*T). 8 waves/block; wave: 32 m-rows x 64 f (8 C-tiles).
// B-fragments are reused across the two m-subtiles -> 1.5 LDS dw per WMMA.
__global__ void k_spat2(const float* __restrict__ S, const float* __restrict__ G,
                        const float* __restrict__ H, const float* __restrict__ dinv,
                        const float* __restrict__ sp, float* __restrict__ Y) {
  __shared__ float sA[256 * SAS];                    // 36 KB (padded, conflict-free)
  __shared__ float sB[32 * 64];                      //  8 KB
  const int mb = blockIdx.x * 256;
  const int bt = blockIdx.y;
  const int t = bt % TT;
  const int tid = threadIdx.x;
  const int wave = tid >> 5, lane = tid & 31;
  const int l15 = lane & 15;
  const int hi = (lane >> 4) << 1;
  const float s0 = fmaxf(sp[0], 0.f), s2 = fmaxf(sp[2], 0.f);
  const float* Gb = G + (size_t)bt * NF;
  v8f acc[2][4];
#pragma unroll
  for (int s = 0; s < 2; ++s)
#pragma unroll
    for (int i = 0; i < 4; ++i) acc[s][i] = {};
  for (int k0 = 0; k0 < NN; k0 += 32) {
    // stage S tile 256x32 (2048 float4, 8 per thread)
#pragma unroll
    for (int i = 0; i < 8; ++i) {
      int e4 = tid + i * 256, r = e4 >> 3, c4 = e4 & 7;
      stage16(&S[(size_t)(mb + r) * NN + k0 + 4 * c4], &sA[r * SAS + 4 * c4]);
    }
    // stage G tile 32x64 (512 float4, 2 per thread)
#pragma unroll
    for (int i = 0; i < 2; ++i) {
      int e4 = tid + i * 256, r = e4 >> 4, c4 = e4 & 15;
      stage16(&Gb[(size_t)(k0 + r) * 64 + 4 * c4], &sB[r * 64 + 4 * c4]);
    }
    stage_wait();
    // prefetch next chunk toward L2 while this one computes
    if (k0 + 32 < NN) {
      __builtin_prefetch(&S[(size_t)(mb + tid) * NN + k0 + 32], 0, 3);
      if (tid < 64)
        __builtin_prefetch(&Gb[(size_t)(k0 + 32 + (tid >> 1)) * 64 + (tid & 1) * 32], 0, 3);
    }
    __syncthreads();
#pragma unroll
    for (int kk = 0; kk < 32; kk += 4) {
      int kb = kk + hi;
      v2f a0, a1;
      a0.x = sA[(wave * 32 + l15) * SAS + kb];
      a0.y = sA[(wave * 32 + l15) * SAS + kb + 1];
      a1.x = sA[(wave * 32 + 16 + l15) * SAS + kb];
      a1.y = sA[(wave * 32 + 16 + l15) * SAS + kb + 1];
#pragma unroll
      for (int ft = 0; ft < 4; ++ft) {
        v2f b;
        b.x = sB[kb * 64 + ft * 16 + l15];
        b.y = sB[(kb + 1) * 64 + ft * 16 + l15];
        acc[0][ft] = wmma4(a0, b, acc[0][ft]);
        acc[1][ft] = wmma4(a1, b, acc[1][ft]);
      }
    }
    __syncthreads();
  }
#pragma unroll
  for (int sub = 0; sub < 2; ++sub) {
    const int mbase = mb + wave * 32 + sub * 16 + ((lane >> 4) << 3);
#pragma unroll
    for (int ft = 0; ft < 4; ++ft) {
      int f = ft * 16 + l15;
#pragma unroll
      for (int r = 0; r < 8; ++r) {
        int m = mbase + r;
        size_t hidx = ((size_t)bt * NN + m) * 64 + f;
        float ht = 0.f;
        if (t >= 1) ht += WD1 * H[hidx - (size_t)NF];
        if (t >= 2) ht += WD2 * H[hidx - (size_t)2 * NF];
        if (t >= 3) ht += WD3 * H[hidx - (size_t)3 * NF];
        Y[hidx] = (acc[sub][ft][r] + s0 * H[hidx] + s2 * ht) * dinv[t * NN + m];
      }
    }
  }
}

// Z[row,g] (init: = bias + ; acc: +=) sum_f Xin[row,f]*W[f,g]. W (64x64) in LDS.
__global__ void k_featgemm(const float* __restrict__ Xin, const float* __restrict__ W,
                           const float* __restrict__ bias, float* __restrict__ Z,
                           int initFlag) {
  __shared__ float sW[64 * 64];
  const int tid = threadIdx.x;
#pragma unroll
  for (int i = 0; i < 16; ++i) sW[tid + i * 256] = W[tid + i * 256];
  __syncthreads();
  const int wave = tid >> 5, lane = tid & 31;
  const int l15 = lane & 15;
  const int hi = (lane >> 4) << 1;
  const int rowb = blockIdx.x * 128 + wave * 16;
  v8f acc[4];
#pragma unroll
  for (int i = 0; i < 4; ++i) acc[i] = {};
#pragma unroll
  for (int kk = 0; kk < 64; kk += 4) {
    int kb = kk + hi;
    int m = rowb + l15;
    v2f a;
    a.x = Xin[(size_t)m * 64 + kb];
    a.y = Xin[(size_t)m * 64 + kb + 1];
#pragma unroll
    for (int ft = 0; ft < 4; ++ft) {
      v2f b;
      b.x = sW[kb * 64 + ft * 16 + l15];
      b.y = sW[(kb + 1) * 64 + ft * 16 + l15];
      acc[ft] = wmma4(a, b, acc[ft]);
    }
  }
  const int mbase = rowb + ((lane >> 4) << 3);
#pragma unroll
  for (int ft = 0; ft < 4; ++ft) {
    int g = ft * 16 + l15;
#pragma unroll
    for (int r = 0; r < 8; ++r) {
      size_t zi = (size_t)(mbase + r) * 64 + g;
      float v = acc[ft][r];
      if (initFlag) Z[zi] = v + bias[g];
      else          Z[zi] += v;
    }
  }
}

// relu + mean over T + head: out[b,m] = bh + sum_f Wh[f]*mean_t relu(Z[b,t,m,f])
__global__ void k_head(const float* __restrict__ Z, const float* __restrict__ Wh,
                       const float* __restrict__ bh, float* __restrict__ out) {
  const int wave = threadIdx.x >> 5, lane = threadIdx.x & 31;
  const int oi = blockIdx.x * 8 + wave;              // < B*N
  const int b = oi >> 11, m = oi & (NN - 1);
  float acc = 0.f;
#pragma unroll
  for (int ff = 0; ff < 2; ++ff) {
    int f = lane + ff * 32;
    float s = 0.f;
#pragma unroll
    for (int t = 0; t < TT; ++t)
      s += fmaxf(Z[(((size_t)(b * TT + t)) * NN + m) * 64 + f], 0.f);
    acc += Wh[f] * s;
  }
#pragma unroll
  for (int off = 16; off >= 1; off >>= 1) acc += __shfl_xor(acc, off, 32);
  if (lane == 0) out[oi] = bh[0] + acc * (1.0f / (float)TT);
}

// ---------------------------------------------------------------------------
extern "C" void kernel_launch(void* const* d_in, const int* in_sizes, int n_in,
                              void* d_out, int out_size, void* d_ws, size_t ws_size,
                              hipStream_t stream) {
  (void)in_sizes; (void)n_in; (void)out_size; (void)ws_size;
  const float* x  = (const float*)d_in[0];
  const float* S  = (const float*)d_in[1];
  const float* sp = (const float*)d_in[2];
  const float* W1 = (const float*)d_in[3];
  const float* b1 = (const float*)d_in[4];
  const float* W2 = (const float*)d_in[5];
  const float* b2 = (const float*)d_in[6];
  const float* Wh = (const float*)d_in[7];
  const float* bh = (const float*)d_in[8];
  float* out = (float*)d_out;

  float* ws  = (float*)d_ws;
  float* rsS = ws;                 // 2048
  float* dnv = rsS + NN;           // T*N = 24576
  float* H0  = dnv + TT * NN;      // NBT each
  float* H1  = H0 + NBT;
  float* H2  = H1 + NBT;
  float* G1  = H2 + NBT;
  float* X   = G1 + NBT;           // BTNF each
  float* X1  = X + BTNF;
  float* GB  = X1 + BTNF;
  float* Z   = GB + BTNF;

  // --- setup ---
  k_transpose_x<<<NBT / 256, 256, 0, stream>>>(x, H0);
  k_rowsum<<<NN, 256, 0, stream>>>(S, rsS);
  k_dinv<<<(TT * NN) / 256, 256, 0, stream>>>(rsS, sp, dnv);

  // --- layer 1 (F=1, 96 columns) ---
  k_tmix1<<<NBT / 256, 256, 0, stream>>>(H0, sp, G1);
  k_spat1<<<dim3(NN / 128, BT / 16), 256, 0, stream>>>(S, G1, H0, dnv, sp, H1);
  k_tmix1<<<NBT / 256, 256, 0, stream>>>(H1, sp, G1);
  k_spat1<<<dim3(NN / 128, BT / 16), 256, 0, stream>>>(S, G1, H1, dnv, sp, H2);
  k_l1outer<<<BTNF / 256, 256, 0, stream>>>(H0, H1, H2, W1, b1, X);

  // --- layer 2 (F=64) ---
  k_featgemm<<<(BT * NN) / 128, 256, 0, stream>>>(X, W2, b2, Z, 1);
  k_tmix2<<<BTNF / 256, 256, 0, stream>>>(X, sp, GB);
  k_spat2<<<dim3(NN / 256, BT), 256, 0, stream>>>(S, GB, X, dnv, sp, X1);
  k_featgemm<<<(BT * NN) / 128, 256, 0, stream>>>(X1, W2 + 64 * 64, nullptr, Z, 0);
  k_tmix2<<<BTNF / 256, 256, 0, stream>>>(X1, sp, GB);
  k_spat2<<<dim3(NN / 256, BT), 256, 0, stream>>>(S, GB, X1, dnv, sp, X); // X2 -> X buf
  k_featgemm<<<(BT * NN) / 128, 256, 0, stream>>>(X, W2 + 2 * 64 * 64, nullptr, Z, 0);

  // --- pool + head ---
  k_head<<<(BB * NN) / 8, 256, 0, stream>>>(Z, Wh, bh, out);
}